// Quantize_10007273800157
// MI455X (gfx1250) — compile-verified
//
#include <hip/hip_runtime.h>
#include <hip/hip_bf16.h>

// Problem constants (from reference):
//   inputs:  [256,32,32,64] f32  -> flat [N=262144, D=64]
//   embedding: [D=64, K=512] f32
#define VQ_N   262144
#define VQ_D   64
#define VQ_K   512
#define VQ_ND  16777216           // N*D
#define EMB_PAD 66                // LDS row pitch for transposed codebook (even -> 8B aligned pairs)

typedef float v2f __attribute__((ext_vector_type(2)));
typedef float v8f __attribute__((ext_vector_type(8)));

// ---------------------------------------------------------------------------
// Kernel A: init workspace — e_norms[k] = sum_d E[d,k]^2, zero accumulators.
// ---------------------------------------------------------------------------
__global__ __launch_bounds__(256) void vq_init(const float* __restrict__ emb,
                                               float* __restrict__ enorm,
                                               float* __restrict__ ws_diff,
                                               float* __restrict__ counts,
                                               float* __restrict__ embsum) {
  int t = threadIdx.x;
  for (int k = t; k < VQ_K; k += 256) {
    float s = 0.f;
    #pragma unroll 8
    for (int d = 0; d < VQ_D; ++d) {
      float e = emb[d * VQ_K + k];
      s += e * e;
    }
    enorm[k]  = s;
    counts[k] = 0.f;
  }
  for (int i = t; i < VQ_D * VQ_K; i += 256) embsum[i] = 0.f;
  if (t == 0) ws_diff[0] = 0.f;
}

// ---------------------------------------------------------------------------
// Kernel B: main VQ kernel. 256 threads = 8 waves; each wave owns 16 rows.
// Distances via V_WMMA_F32_16X16X4_F32 against LDS-transposed codebook,
// per-lane argmin + wave32 shfl_xor butterfly, then quantize gather + MSE.
// Dynamic LDS: emb_t[512*66] | enorm_s[512] | idx_s[128] (int) | red[8]
// ---------------------------------------------------------------------------
__global__ __launch_bounds__(256) void vq_main(const float* __restrict__ flat,
                                               const float* __restrict__ emb,
                                               const float* __restrict__ enorm,
                                               float* __restrict__ out_quant,
                                               float* __restrict__ out_idx,
                                               float* __restrict__ ws_diff) {
  extern __shared__ float smem[];
  float* emb_t   = smem;                                   // [K][EMB_PAD]
  float* enorm_s = smem + VQ_K * EMB_PAD;                  // [K]
  int*   idx_s   = (int*)(smem + VQ_K * EMB_PAD + VQ_K);   // [128]
  float* red_s   = smem + VQ_K * EMB_PAD + VQ_K + 128;     // [8]

  const int t = threadIdx.x;

  // Stage codebook transposed: emb_t[k][d] = E[d][k] (coalesced global reads,
  // conflict-free LDS writes thanks to pitch 66).
  for (int i = t; i < VQ_D * VQ_K; i += 256) {
    int d = i >> 9;          // /512
    int k = i & (VQ_K - 1);
    emb_t[k * EMB_PAD + d] = emb[i];
  }
  for (int i = t; i < VQ_K; i += 256) enorm_s[i] = enorm[i];
  __syncthreads();

  const int lane = t & 31;
  const int w    = t >> 5;
  const int m    = lane & 15;   // M index within 16-row strip / N within tile
  const int h    = lane >> 4;   // half-wave selector
  const int row0 = blockIdx.x * 128 + w * 16;

  // A-tile: 16 rows x 64 dims, f32 16x16x4 layout -> lane holds d = 4c + 2h + {0,1}
  v2f a[16];
  const float* arow = flat + (size_t)(row0 + m) * VQ_D + 2 * h;
  #pragma unroll
  for (int c = 0; c < 16; ++c)
    a[c] = *(const v2f*)(arow + 4 * c);

  float bestV[8];
  int   bestI[8];
  #pragma unroll
  for (int r = 0; r < 8; ++r) { bestV[r] = 3.4e38f; bestI[r] = 0; }

  // Sweep 32 chunks of 16 codes.
  for (int n = 0; n < 32; ++n) {
    v8f acc = {0.f, 0.f, 0.f, 0.f, 0.f, 0.f, 0.f, 0.f};
    const float* bb = emb_t + (n * 16 + m) * EMB_PAD + 2 * h;
    #pragma unroll
    for (int c = 0; c < 16; ++c) {
      v2f b = *(const v2f*)(bb + 4 * c);
      acc = __builtin_amdgcn_wmma_f32_16x16x4_f32(
          false, a[c], false, b, (short)0, acc, false, false);
    }
    const int   col = n * 16 + m;
    const float en  = enorm_s[col];
    #pragma unroll
    for (int r = 0; r < 8; ++r) {
      float s = en - 2.0f * acc[r];   // ||e||^2 - 2 x.e  (row const dropped)
      if (s < bestV[r]) { bestV[r] = s; bestI[r] = col; }
    }
  }

  // Argmin across the 16 lanes holding the same row (xor stays within half).
  #pragma unroll
  for (int off = 1; off < 16; off <<= 1) {
    #pragma unroll
    for (int r = 0; r < 8; ++r) {
      float ov = __shfl_xor(bestV[r], off, 32);
      int   oi = __shfl_xor(bestI[r], off, 32);
      if (ov < bestV[r] || (ov == bestV[r] && oi < bestI[r])) {
        bestV[r] = ov; bestI[r] = oi;
      }
    }
  }

  if (m == 0) {  // lane 0 -> rows 0..7, lane 16 -> rows 8..15
    const int rb = h * 8;
    #pragma unroll
    for (int r = 0; r < 8; ++r) {
      idx_s[w * 16 + rb + r] = bestI[r];
      out_idx[row0 + rb + r] = (float)bestI[r];
    }
  }
  __syncthreads();

  // Quantize gather + squared-error accumulation: 2 threads per row, 32 dims each.
  float local = 0.f;
  {
    const int rl  = t >> 1;
    const int row = blockIdx.x * 128 + rl;
    const int k   = idx_s[rl];
    const int d0  = (t & 1) * 32;
    const float* xr = flat + (size_t)row * VQ_D + d0;
    float*       qr = out_quant + (size_t)row * VQ_D + d0;
    const float* er = emb_t + k * EMB_PAD + d0;
    #pragma unroll
    for (int i = 0; i < 8; ++i) {
      float4 q;
      q.x = er[4 * i + 0]; q.y = er[4 * i + 1];
      q.z = er[4 * i + 2]; q.w = er[4 * i + 3];
      float4 x = *(const float4*)(xr + 4 * i);
      *(float4*)(qr + 4 * i) = q;
      float dx = q.x - x.x, dy = q.y - x.y, dz = q.z - x.z, dw = q.w - x.w;
      local += dx * dx + dy * dy + dz * dz + dw * dw;
    }
  }
  #pragma unroll
  for (int off = 16; off > 0; off >>= 1) local += __shfl_xor(local, off, 32);
  if (lane == 0) red_s[w] = local;
  __syncthreads();
  if (t == 0) {
    float s = 0.f;
    #pragma unroll
    for (int i = 0; i < 8; ++i) s += red_s[i];
    atomicAdd(ws_diff, s);
  }
}

// ---------------------------------------------------------------------------
// Kernel C: segment sums. 64 blocks, each a contiguous 4096-row slice; LDS scoreboard
// (ds_add_f32 atomics) then a single global-atomic flush per block.
// Dynamic LDS: embsum_s[512*66] | counts_s[512]
// ---------------------------------------------------------------------------
__global__ __launch_bounds__(256) void vq_segsum(const float* __restrict__ flat,
                                                 const float* __restrict__ idx_f,
                                                 float* __restrict__ counts_g,
                                                 float* __restrict__ embsum_g) {
  extern __shared__ float smem[];
  float* es = smem;                      // [K][EMB_PAD]
  float* cs = smem + VQ_K * EMB_PAD;     // [K]

  const int t = threadIdx.x;
  for (int i = t; i < VQ_K * EMB_PAD; i += 256) es[i] = 0.f;
  for (int i = t; i < VQ_K; i += 256) cs[i] = 0.f;
  __syncthreads();

  const int rowsPerBlock = VQ_N / 64;    // 4096
  const int rstart = blockIdx.x * rowsPerBlock;
  const int rl = t >> 1;
  const int d0 = (t & 1) * 32;

  for (int it = 0; it < rowsPerBlock / 128; ++it) {   // 32 iterations
    const int row = rstart + it * 128 + rl;
    const int k   = (int)idx_f[row];
    const float* xr = flat + (size_t)row * VQ_D + d0;
    float* ep = es + k * EMB_PAD + d0;
    #pragma unroll
    for (int d = 0; d < 32; ++d) atomicAdd(&ep[d], xr[d]);
    if (d0 == 0) atomicAdd(&cs[k], 1.0f);
  }
  __syncthreads();

  for (int i = t; i < VQ_D * VQ_K; i += 256) {
    int k = i >> 6, d = i & 63;
    float v = es[k * EMB_PAD + d];
    if (v != 0.f) atomicAdd(&embsum_g[d * VQ_K + k], v);
  }
  for (int i = t; i < VQ_K; i += 256) {
    float v = cs[i];
    if (v != 0.f) atomicAdd(&counts_g[i], v);
  }
}

// ---------------------------------------------------------------------------
// Kernel D: EMA update + Laplace smoothing + diff normalization. 1 block.
// ---------------------------------------------------------------------------
__global__ __launch_bounds__(256) void vq_finalize(const float* __restrict__ cluster_size,
                                                   const float* __restrict__ embeddings_avg,
                                                   const float* __restrict__ ws_diff,
                                                   const float* __restrict__ counts,
                                                   const float* __restrict__ embsum,
                                                   float* __restrict__ out_diff,
                                                   float* __restrict__ out_emb,
                                                   float* __restrict__ out_cs,
                                                   float* __restrict__ out_avg) {
  __shared__ float csn[VQ_K];
  __shared__ float red[256];
  const int t = threadIdx.x;

  float p = 0.f;
  for (int k = t; k < VQ_K; k += 256) {
    float v = 0.99f * cluster_size[k] + 0.01f * counts[k];
    csn[k]    = v;
    out_cs[k] = v;
    p += v;
  }
  red[t] = p;
  __syncthreads();
  for (int s = 128; s > 0; s >>= 1) {
    if (t < s) red[t] += red[t + s];
    __syncthreads();
  }
  const float nsum = red[0];

  for (int k = t; k < VQ_K; k += 256) {
    float c   = (csn[k] + 1e-5f) / (nsum + (float)VQ_K * 1e-5f) * nsum;
    float inv = 1.0f / c;
    #pragma unroll 4
    for (int d = 0; d < VQ_D; ++d) {
      float av = 0.99f * embeddings_avg[d * VQ_K + k] + 0.01f * embsum[d * VQ_K + k];
      out_avg[d * VQ_K + k] = av;
      out_emb[d * VQ_K + k] = av * inv;
    }
  }
  if (t == 0) out_diff[0] = ws_diff[0] * (1.0f / (float)VQ_ND);
}

// ---------------------------------------------------------------------------
extern "C" void kernel_launch(void* const* d_in, const int* in_sizes, int n_in,
                              void* d_out, int out_size, void* d_ws, size_t ws_size,
                              hipStream_t stream) {
  (void)in_sizes; (void)n_in; (void)out_size; (void)ws_size;

  const float* flat      = (const float*)d_in[0];  // [N, D]
  const float* emb       = (const float*)d_in[1];  // [D, K]
  const float* clus_sz   = (const float*)d_in[2];  // [K]
  const float* emb_avg   = (const float*)d_in[3];  // [D, K]

  float* out = (float*)d_out;
  // Output tuple flattened in return order:
  const size_t o_quant  = 0;                        // N*D
  const size_t o_diff   = o_quant + (size_t)VQ_ND;  // 1
  const size_t o_idx    = o_diff + 1;               // N
  const size_t o_newemb = o_idx + (size_t)VQ_N;     // D*K
  const size_t o_newcs  = o_newemb + VQ_D * VQ_K;   // K
  const size_t o_newavg = o_newcs + VQ_K;           // D*K

  // Workspace layout (floats): diff | enorm[K] | counts[K] | embsum[D*K]
  float* ws        = (float*)d_ws;
  float* ws_diff   = ws;
  float* ws_enorm  = ws + 1;
  float* ws_counts = ws + 1 + VQ_K;
  float* ws_embsum = ws + 1 + 2 * VQ_K;

  const size_t lds_main = (VQ_K * EMB_PAD + VQ_K + 128 + 8) * sizeof(float);
  const size_t lds_seg  = (VQ_K * EMB_PAD + VQ_K) * sizeof(float);

  vq_init<<<1, 256, 0, stream>>>(emb, ws_enorm, ws_diff, ws_counts, ws_embsum);
  vq_main<<<VQ_N / 128, 256, lds_main, stream>>>(
      flat, emb, ws_enorm, out + o_quant, out + o_idx, ws_diff);
  vq_segsum<<<64, 256, lds_seg, stream>>>(
      flat, out + o_idx, ws_counts, ws_embsum);
  vq_finalize<<<1, 256, 0, stream>>>(
      clus_sz, emb_avg, ws_diff, ws_counts, ws_embsum,
      out + o_diff, out + o_newemb, out + o_newcs, out + o_newavg);
}